// RetinaLoss_44676249813185
// MI455X (gfx1250) — compile-verified
//
#include <hip/hip_runtime.h>
#include <cstdint>

typedef float v2f __attribute__((ext_vector_type(2)));
typedef float v8f __attribute__((ext_vector_type(8)));
typedef float f4v __attribute__((ext_vector_type(4)));

constexpr int   B = 8, A = 76725, C = 80, M = 20;
constexpr float IMAGE_W = 640.0f, IMAGE_H = 640.0f;
constexpr float ALPHA = 0.25f, BETAC = 1.0f / 9.0f, EPSV = 1.0e-4f;

constexpr int BLK       = 256;                        // 8 waves (wave32)
constexpr int ITEMS     = 4;                          // float4 chunks / thread
constexpr int TILE      = BLK * ITEMS;                // 1024 chunks / block
constexpr int CH4       = (A * C) / 4;                // float4 chunks per image
constexpr int NB_FOCAL  = (CH4 + TILE - 1) / TILE;    // focal blocks per image
constexpr int NB_ASSIGN = (A + BLK - 1) / BLK;        // assign blocks per image
static_assert((A * C) % 4 == 0, "C chunking");

// ---------------------------------------------------------------------------
// Wave-level sum via V_WMMA_F32_16X16X4_F32:
//   A(16x4) = ones, B(4x16) carries the 32 lane values in its first VGPR
//   (second VGPR zero). D[m][n] = sum_k B[k][n]  => 16 column sums live in
//   lanes 0..15 of c[0] (mirrored in 16..31). 4 xor-shuffles finish (the
//   N-indexed sums would need a lane transpose to chain another WMMA).
// Exact f32 accumulation; requires EXEC == all ones (callers guarantee it).
// ---------------------------------------------------------------------------
__device__ __forceinline__ float waveReduceWmma(float v) {
    v2f a; a[0] = 1.0f; a[1] = 1.0f;
    v2f bm; bm[0] = v; bm[1] = 0.0f;
    v8f c = {};
    c = __builtin_amdgcn_wmma_f32_16x16x4_f32(false, a, false, bm,
                                              (short)0, c, false, false);
    float s = c[0];
    s += __shfl_xor(s, 1, 32);
    s += __shfl_xor(s, 2, 32);
    s += __shfl_xor(s, 4, 32);
    s += __shfl_xor(s, 8, 32);
    return s;                       // wave total in every lane
}

// Deterministic block reduction (fixed tree order). Result valid on thread 0.
__device__ __forceinline__ float blockReduce(float v, float* smem) {
    const int lane = threadIdx.x & 31;
    const int wid  = threadIdx.x >> 5;
    const float w = waveReduceWmma(v);
    if (lane == 0) smem[wid] = w;
    __syncthreads();
    float r = 0.0f;
    if (threadIdx.x == 0) {
        const int nw = blockDim.x >> 5;
        for (int i = 0; i < nw; ++i) r += smem[i];
    }
    __syncthreads();                // make smem reusable for the next call
    return r;
}

// ---------------------------------------------------------------------------
// Kernel 1: anchor assignment + smooth-L1 reg loss (fused).
// grid = (NB_ASSIGN, B). Writes per-anchor cls label and per-block partials.
// ---------------------------------------------------------------------------
__global__ void assign_reg_kernel(const float* __restrict__ reg_heads,
                                  const float* __restrict__ anchors,
                                  const float* __restrict__ ann,
                                  float* __restrict__ cls_t,
                                  float* __restrict__ reg_part,
                                  float* __restrict__ pos_part) {
    __shared__ float s_ann[M * 5];
    __shared__ float s_red[BLK / 32];
    const int b   = blockIdx.y;
    const int tid = threadIdx.x;
    if (tid < M * 5) s_ann[tid] = ann[b * M * 5 + tid];
    __syncthreads();

    const int aidx = blockIdx.x * blockDim.x + tid;
    float slp = 0.0f, posf = 0.0f;
    if (aidx < A) {
        const size_t base = ((size_t)b * A + aidx) * 4;
        const f4v an = *(const f4v*)(anchors + base);
        const float ax0 = an[0], ay0 = an[1], ax1 = an[2], ay1 = an[3];
        const bool border = (ax0 > 0.0f) && (ay0 > 0.0f) &&
                            (ax1 < IMAGE_W) && (ay1 < IMAGE_H);
        const float area_a = (ax1 - ax0) * (ay1 - ay0);

        float best = -1.0f; int bi = 0; bool anyv = false;
        for (int m = 0; m < M; ++m) {
            const float g0 = s_ann[m * 5 + 0], g1 = s_ann[m * 5 + 1];
            const float g2 = s_ann[m * 5 + 2], g3 = s_ann[m * 5 + 3];
            const float gc = s_ann[m * 5 + 4];
            float iou = -1.0f;
            if (gc >= 0.0f) {
                anyv = true;
                float w = fminf(ax1, g2) - fmaxf(ax0, g0);
                float h = fminf(ay1, g3) - fmaxf(ay0, g1);
                w = fmaxf(w, 0.0f); h = fmaxf(h, 0.0f);
                const float inter = w * h;
                const float uni = area_a + (g2 - g0) * (g3 - g1) - inter;
                iou = inter / fmaxf(uni, 1.0e-6f);
            }
            if (iou > best) { best = iou; bi = m; }  // first-max tie-break
        }

        float cls = -1.0f;
        if (anyv && border) {
            if (best < 0.4f)       cls = 0.0f;
            else if (best >= 0.5f) cls = s_ann[bi * 5 + 4] + 1.0f;
        }
        cls_t[(size_t)b * A + aidx] = cls;

        if (cls > 0.0f) {                      // positive: reg target + SL1
            posf = 1.0f;
            const float g0 = s_ann[bi * 5 + 0], g1 = s_ann[bi * 5 + 1];
            const float g2 = s_ann[bi * 5 + 2], g3 = s_ann[bi * 5 + 3];
            const float awx = ax1 - ax0, awy = ay1 - ay0;
            const float acx = ax0 + 0.5f * awx, acy = ay0 + 0.5f * awy;
            const float gwx = fmaxf(g2 - g0, 1.0f), gwy = fmaxf(g3 - g1, 1.0f);
            const float gcx = g0 + 0.5f * gwx, gcy = g1 + 0.5f * gwy;
            float t[4];
            t[0] = ((gcx - acx) / awx) / 0.1f;
            t[1] = ((gcy - acy) / awy) / 0.1f;
            t[2] = __logf(gwx / awx) / 0.2f;
            t[3] = __logf(gwy / awy) / 0.2f;
            const f4v rh = *(const f4v*)(reg_heads + base);
            float s4 = 0.0f;
#pragma unroll
            for (int i = 0; i < 4; ++i) {
                const float x = fabsf(rh[i] - t[i]);
                s4 += (x >= BETAC) ? (x - 0.5f * BETAC)
                                   : (0.5f * x * x / BETAC);
            }
            slp = 0.25f * s4;                  // mean over 4 coords
        }
    }
    const float rsum = blockReduce(slp, s_red);
    const float psum = blockReduce(posf, s_red);
    if (tid == 0) {
        reg_part[b * gridDim.x + blockIdx.x] = rsum;
        pos_part[b * gridDim.x + blockIdx.x] = psum;
    }
}

// ---------------------------------------------------------------------------
// Kernel 2: focal loss, the 196 MB streaming pass. grid = (NB_FOCAL, B).
// Each thread handles ITEMS float4 chunks of cls_heads, block-contiguous so
// lanes stay fully coalesced; loads are non-temporal (single-use stream,
// don't thrash L2). Rows with cls < 0 (keep==0) are never loaded.
// ---------------------------------------------------------------------------
__global__ void focal_kernel(const float* __restrict__ cls_heads,
                             const float* __restrict__ cls_t,
                             float* __restrict__ cls_part) {
    __shared__ float s_red[BLK / 32];
    const int b    = blockIdx.y;
    const int base = blockIdx.x * TILE + threadIdx.x;
    float s = 0.0f;
#pragma unroll
    for (int k = 0; k < ITEMS; ++k) {
        const int idx = base + k * BLK;
        if (idx < CH4) {
            const int anchor = idx / (C / 4);
            const int c0     = (idx % (C / 4)) * 4;
            const float cl   = cls_t[(size_t)b * A + anchor];
            if (cl >= 0.0f) {                   // keep
                const int lab = (int)cl - 1;    // -1 => background
                const f4v h = __builtin_nontemporal_load(
                    (const f4v*)cls_heads + (size_t)b * CH4 + idx);
#pragma unroll
                for (int j = 0; j < 4; ++j) {
                    float p = fminf(fmaxf(h[j], EPSV), 1.0f - EPSV);
                    if (c0 + j == lab) {
                        const float q = 1.0f - p;
                        s += ALPHA * q * q * (-__logf(p));
                    } else {
                        s += (1.0f - ALPHA) * p * p * (-__logf(1.0f - p));
                    }
                }
            }
        }
    }
    const float t = blockReduce(s, s_red);
    if (threadIdx.x == 0) cls_part[b * gridDim.x + blockIdx.x] = t;
}

// ---------------------------------------------------------------------------
// Kernel 3: per-image deterministic reduction of all partials.
// grid = (B). per_img layout: [cls_raw(B) | reg_raw(B) | pos(B)]
// ---------------------------------------------------------------------------
__global__ void finalize_img_kernel(const float* __restrict__ cls_part,
                                    const float* __restrict__ reg_part,
                                    const float* __restrict__ pos_part,
                                    float* __restrict__ per_img) {
    __shared__ float s_red[BLK / 32];
    const int b   = blockIdx.x;
    const int tid = threadIdx.x;
    float cs = 0.0f, rs = 0.0f, ps = 0.0f;
    for (int i = tid; i < NB_FOCAL; i += blockDim.x)
        cs += cls_part[b * NB_FOCAL + i];
    for (int i = tid; i < NB_ASSIGN; i += blockDim.x) {
        rs += reg_part[b * NB_ASSIGN + i];
        ps += pos_part[b * NB_ASSIGN + i];
    }
    const float cst = blockReduce(cs, s_red);
    const float rst = blockReduce(rs, s_red);
    const float pst = blockReduce(ps, s_red);
    if (tid == 0) {
        per_img[b]         = cst;
        per_img[B + b]     = rst;
        per_img[2 * B + b] = pst;
    }
}

// Kernel 4: combine 8 images -> 2 scalars. Single wave, fixed order.
__global__ void final_kernel(const float* __restrict__ per_img,
                             float* __restrict__ out) {
    const int lane = threadIdx.x;
    float cls_v = 0.0f, reg_v = 0.0f, val = 0.0f;
    if (lane < B) {
        const float pos   = per_img[2 * B + lane];
        const bool  valid = pos > 0.0f;
        const float safe  = fmaxf(pos, 1.0f);
        cls_v = valid ? per_img[lane] / safe     : 0.0f;
        reg_v = valid ? per_img[B + lane] / safe : 0.0f;
        val   = valid ? 1.0f : 0.0f;
    }
    for (int m = 16; m > 0; m >>= 1) {
        cls_v += __shfl_xor(cls_v, m, 32);
        reg_v += __shfl_xor(reg_v, m, 32);
        val   += __shfl_xor(val,   m, 32);
    }
    if (lane == 0) {
        const float n = fmaxf(val, 1.0f);
        out[0] = cls_v / n;
        out[1] = reg_v / n;
    }
}

// ---------------------------------------------------------------------------
extern "C" void kernel_launch(void* const* d_in, const int* in_sizes, int n_in,
                              void* d_out, int out_size, void* d_ws,
                              size_t ws_size, hipStream_t stream) {
    (void)in_sizes; (void)n_in; (void)out_size; (void)ws_size;
    const float* cls_heads = (const float*)d_in[0];   // (B,A,C)
    const float* reg_heads = (const float*)d_in[1];   // (B,A,4)
    const float* anchors   = (const float*)d_in[2];   // (B,A,4)
    const float* ann       = (const float*)d_in[3];   // (B,M,5)

    float* ws       = (float*)d_ws;                   // ~2.6 MB used
    float* cls_t    = ws;                             // B*A
    float* cls_part = cls_t    + (size_t)B * A;       // B*NB_FOCAL
    float* reg_part = cls_part + (size_t)B * NB_FOCAL;// B*NB_ASSIGN
    float* pos_part = reg_part + (size_t)B * NB_ASSIGN;
    float* per_img  = pos_part + (size_t)B * NB_ASSIGN; // 3*B

    dim3 g1(NB_ASSIGN, B);
    assign_reg_kernel<<<g1, BLK, 0, stream>>>(reg_heads, anchors, ann,
                                              cls_t, reg_part, pos_part);
    dim3 g2(NB_FOCAL, B);
    focal_kernel<<<g2, BLK, 0, stream>>>(cls_heads, cls_t, cls_part);
    finalize_img_kernel<<<B, BLK, 0, stream>>>(cls_part, reg_part, pos_part,
                                               per_img);
    final_kernel<<<1, 32, 0, stream>>>(per_img, (float*)d_out);
}